// SoftclDiceFocalLoss_57337813402231
// MI455X (gfx1250) — compile-verified
//
#include <hip/hip_runtime.h>
#include <stdint.h>

// ---------------- problem geometry (fixed by the reference) ----------------
#define IMG_H 512
#define IMG_W 512
#define TILE  52
#define HALO  5                     // = dependency depth of soft_skel(ITER=3)
#define LW    (TILE + 2*HALO)       // 62
#define LSZ   (LW*LW)               // 3844
#define NTPB  256
#define NTX   ((IMG_W + TILE - 1)/TILE)   // 10
#define NTY   ((IMG_H + TILE - 1)/TILE)   // 10

// ---------------- CDNA5 async global->LDS path ----------------
#if defined(__has_builtin)
# if __has_builtin(__builtin_amdgcn_global_load_async_to_lds_b32)
#  define HAVE_ASYNC_LDS 1
# endif
# if __has_builtin(__builtin_amdgcn_s_wait_asynccnt)
#  define HAVE_WAIT_ASYNC 1
# endif
#endif
#ifndef HAVE_ASYNC_LDS
# define HAVE_ASYNC_LDS 0
#endif
#ifndef HAVE_WAIT_ASYNC
# define HAVE_WAIT_ASYNC 0
#endif

// Builtin signature (per hipcc diagnostic): arg0 = addrspace(1) int*  (global),
// arg1 = addrspace(3) int* (LDS), then two immediate ints (offset, cpol).
typedef __attribute__((address_space(1))) int gbl_i32_t;
typedef __attribute__((address_space(3))) int lds_i32_t;

// Copy one 32-bit word global->LDS via the CDNA5 async-copy path.
// AS casts: flat global == AS1 bit pattern; flat LDS pointer low 32 bits are
// exactly the AS3 offset (clang builds flat LDS ptrs as {aperture, as3}).
__device__ __forceinline__ void g2l_async_b32(const void* g, void* l) {
#if HAVE_ASYNC_LDS
  __builtin_amdgcn_global_load_async_to_lds_b32(
      (gbl_i32_t*)(uintptr_t)g,
      (lds_i32_t*)(uint32_t)(uintptr_t)l,
      0, 0);
#else
  *(int*)l = *(const int*)g;      // synchronous fallback
#endif
}

__device__ __forceinline__ void async_wait_all() {
#if HAVE_ASYNC_LDS
# if HAVE_WAIT_ASYNC
  __builtin_amdgcn_s_wait_asynccnt(0);
# else
  asm volatile("s_wait_asynccnt 0" ::: "memory");
# endif
#endif
}

// ---------------- LDS stencil stages ----------------
// soft_erode = min over the 5-point cross.  Out-of-image output cells get
// `padv` (1.0 if the buffer feeds another erode, 0.0 if it feeds a dilate),
// which reproduces the reference's +inf/-inf reduce_window padding exactly
// since all values are in [0,1] and the window contains the center pixel.
__device__ __forceinline__ void erode_tile(const float* __restrict__ s,
                                           float* __restrict__ d,
                                           int by0, int bx0, float padv, int tid)
{
  for (int idx = tid; idx < LSZ; idx += NTPB) {
    int ly = idx / LW;
    int lx = idx - ly * LW;
    if (ly == 0 || ly == LW-1 || lx == 0 || lx == LW-1) continue;
    int gy = by0 + ly, gx = bx0 + lx;
    float v;
    if ((unsigned)gy < (unsigned)IMG_H && (unsigned)gx < (unsigned)IMG_W) {
      float c    = s[idx];
      float vert = fminf(fminf(s[idx-LW], s[idx+LW]), c);
      float horz = fminf(fminf(s[idx-1],  s[idx+1]),  c);
      v = fminf(vert, horz);
    } else {
      v = padv;
    }
    d[idx] = v;
  }
}

// soft_dilate = 3x3 max.  Its output is only consumed pointwise at in-image
// pixels, so no coordinate handling needed (pads live in the *input* buffer).
__device__ __forceinline__ void dilate_tile(const float* __restrict__ s,
                                            float* __restrict__ d, int tid)
{
  for (int idx = tid; idx < LSZ; idx += NTPB) {
    int ly = idx / LW;
    int lx = idx - ly * LW;
    if (ly == 0 || ly == LW-1 || lx == 0 || lx == LW-1) continue;
    float m = s[idx];
    m = fmaxf(m, s[idx-1]);     m = fmaxf(m, s[idx+1]);
    m = fmaxf(m, s[idx-LW]);    m = fmaxf(m, s[idx+LW]);
    m = fmaxf(m, s[idx-LW-1]);  m = fmaxf(m, s[idx-LW+1]);
    m = fmaxf(m, s[idx+LW-1]);  m = fmaxf(m, s[idx+LW+1]);
    d[idx] = m;
  }
}

__device__ __forceinline__ bool interior_in_image(int idx, int by0, int bx0,
                                                  int& gy, int& gx)
{
  int ly = idx / LW;
  int lx = idx - ly * LW;
  if (ly < HALO || ly >= HALO+TILE || lx < HALO || lx >= HALO+TILE) return false;
  gy = by0 + ly; gx = bx0 + lx;
  return (unsigned)gy < (unsigned)IMG_H && (unsigned)gx < (unsigned)IMG_W;
}

__device__ __forceinline__ void skel_init(const float* __restrict__ img,
                                          const float* __restrict__ opn,
                                          float* __restrict__ S,
                                          int by0, int bx0, int tid)
{
  for (int idx = tid; idx < LSZ; idx += NTPB) {
    int gy, gx;
    if (!interior_in_image(idx, by0, bx0, gy, gx)) continue;
    S[idx] = fmaxf(img[idx] - opn[idx], 0.0f);
  }
}

__device__ __forceinline__ void skel_update(const float* __restrict__ img,
                                            const float* __restrict__ opn,
                                            float* __restrict__ S,
                                            int by0, int bx0, int tid)
{
  for (int idx = tid; idx < LSZ; idx += NTPB) {
    int gy, gx;
    if (!interior_in_image(idx, by0, bx0, gy, gx)) continue;
    float delta = fmaxf(img[idx] - opn[idx], 0.0f);
    float s     = S[idx];
    S[idx] = s + fmaxf(delta - s * delta, 0.0f);
  }
}

// Full soft_skel (ITER=3) over one tile. On entry pa holds the image with
// out-of-image cells preset to 1.0. Clobbers pa/pb/pc; result in S (interior).
__device__ void soft_skel_tile(float* pa, float* pb, float* pc, float* S,
                               int by0, int bx0, int tid)
{
  // img1 = open(img)
  erode_tile (pa, pb, by0, bx0, 0.0f, tid); __syncthreads();
  dilate_tile(pb, pc, tid);                 __syncthreads();
  skel_init  (pa, pc, S, by0, bx0, tid);    __syncthreads();

  #pragma unroll 1
  for (int it = 0; it < 3; ++it) {
    erode_tile (pa, pb, by0, bx0, 1.0f, tid); __syncthreads(); // img = erode(img)
    erode_tile (pb, pc, by0, bx0, 0.0f, tid); __syncthreads(); // open: inner erode
    dilate_tile(pc, pa, tid);                 __syncthreads(); // open: dilate
    skel_update(pb, pa, S, by0, bx0, tid);    __syncthreads();
    float* t = pa; pa = pb; pb = pc; pc = t;                   // img <- pb
  }
}

__device__ __forceinline__ float block_reduce(float v, float* red, int tid)
{
  __syncthreads();
  red[tid] = v;
  __syncthreads();
  for (int s = NTPB >> 1; s > 0; s >>= 1) {
    if (tid < s) red[tid] += red[tid + s];
    __syncthreads();
  }
  return red[0];
}

// ---------------- kernel 1: fused focal + double soft_skel per tile ----------------
__global__ __launch_bounds__(NTPB)
void cl_dice_focal_tile_kernel(const float* __restrict__ logits,
                               const int*   __restrict__ targets,
                               float*       __restrict__ ws)
{
  __shared__ float P0[LSZ];
  __shared__ float P1[LSZ];
  __shared__ float P2[LSZ];
  __shared__ float S [LSZ];
  __shared__ float red[NTPB];

  const int tid = threadIdx.x;
  const int b   = blockIdx.z;
  const int by0 = blockIdx.y * TILE - HALO;
  const int bx0 = blockIdx.x * TILE - HALO;
  const float* img_base = logits  + (size_t)b * (IMG_H * IMG_W);
  const int*   tgt_base = targets + (size_t)b * (IMG_H * IMG_W);

  // warm L2 for this tile's rows
  {
    int cy = min(max(by0, 0), IMG_H - 1), cx = min(max(bx0, 0), IMG_W - 1);
    __builtin_prefetch(img_base + cy * IMG_W + cx, 0, 1);
    __builtin_prefetch(tgt_base + cy * IMG_W + cx, 0, 1);
  }

  // ---------- pass 1: probabilities ----------
  for (int idx = tid; idx < LSZ; idx += NTPB) {
    int ly = idx / LW, lx = idx - ly * LW;
    int cy = min(max(by0 + ly, 0), IMG_H - 1);
    int cx = min(max(bx0 + lx, 0), IMG_W - 1);
    g2l_async_b32(img_base + cy * IMG_W + cx, &P0[idx]);
  }
  async_wait_all();
  __syncthreads();

  // convert logits -> sigmoid in LDS, fuse focal loss over interior pixels
  float facc = 0.0f;
  for (int idx = tid; idx < LSZ; idx += NTPB) {
    int ly = idx / LW, lx = idx - ly * LW;
    int gy = by0 + ly, gx = bx0 + lx;
    bool in = (unsigned)gy < (unsigned)IMG_H && (unsigned)gx < (unsigned)IMG_W;
    float x = P0[idx];
    float p = 1.0f / (1.0f + __expf(-x));
    if (in) {
      P0[idx] = p;
      if (ly >= HALO && ly < HALO+TILE && lx >= HALO && lx < HALO+TILE) {
        float t  = (float)tgt_base[gy * IMG_W + gx];
        float pt = p * t + (1.0f - p) * (1.0f - t);
        float af = t * 0.01f + (1.0f - t) * 0.99f;
        float om = 1.0f - pt;
        facc -= af * om * om * __logf(fmaxf(pt, 1e-6f));
      }
    } else {
      P0[idx] = 1.0f;             // pad for the first erode
    }
  }
  __syncthreads();

  soft_skel_tile(P0, P1, P2, S, by0, bx0, tid);

  float sp = 0.0f, spt = 0.0f;
  for (int idx = tid; idx < LSZ; idx += NTPB) {
    int gy, gx;
    if (!interior_in_image(idx, by0, bx0, gy, gx)) continue;
    float s = S[idx];
    sp  += s;
    spt += s * (float)tgt_base[gy * IMG_W + gx];
  }
  __syncthreads();

  // ---------- pass 2: targets ----------
  for (int idx = tid; idx < LSZ; idx += NTPB) {
    int ly = idx / LW, lx = idx - ly * LW;
    int cy = min(max(by0 + ly, 0), IMG_H - 1);
    int cx = min(max(bx0 + lx, 0), IMG_W - 1);
    g2l_async_b32(tgt_base + cy * IMG_W + cx, &P0[idx]);   // raw i32 bits
  }
  async_wait_all();
  __syncthreads();

  for (int idx = tid; idx < LSZ; idx += NTPB) {
    int ly = idx / LW, lx = idx - ly * LW;
    int gy = by0 + ly, gx = bx0 + lx;
    bool in = (unsigned)gy < (unsigned)IMG_H && (unsigned)gx < (unsigned)IMG_W;
    int bits = __float_as_int(P0[idx]);
    P0[idx] = in ? (float)bits : 1.0f;
  }
  __syncthreads();

  soft_skel_tile(P0, P1, P2, S, by0, bx0, tid);

  float st = 0.0f, stq = 0.0f;
  for (int idx = tid; idx < LSZ; idx += NTPB) {
    int gy, gx;
    if (!interior_in_image(idx, by0, bx0, gy, gx)) continue;
    float s = S[idx];
    st  += s;
    float x = img_base[gy * IMG_W + gx];
    stq += s * (1.0f / (1.0f + __expf(-x)));
  }

  // ---------- deterministic per-block partials ----------
  const int nblk = gridDim.x * gridDim.y * gridDim.z;
  const int bid  = (blockIdx.z * gridDim.y + blockIdx.y) * gridDim.x + blockIdx.x;
  float r0 = block_reduce(facc, red, tid);
  float r1 = block_reduce(sp,   red, tid);
  float r2 = block_reduce(spt,  red, tid);
  float r3 = block_reduce(st,   red, tid);
  float r4 = block_reduce(stq,  red, tid);
  if (tid == 0) {
    ws[0 * nblk + bid] = r0;
    ws[1 * nblk + bid] = r1;
    ws[2 * nblk + bid] = r2;
    ws[3 * nblk + bid] = r3;
    ws[4 * nblk + bid] = r4;
  }
}

// ---------------- kernel 2: final reduction + scalar combine ----------------
__global__ __launch_bounds__(NTPB)
void cl_dice_focal_final_kernel(const float* __restrict__ ws, int nblk,
                                float inv_n, float* __restrict__ out)
{
  __shared__ float red[NTPB];
  const int tid = threadIdx.x;
  float a[5];
  for (int j = 0; j < 5; ++j) {
    float v = 0.0f;
    for (int i = tid; i < nblk; i += NTPB) v += ws[(size_t)j * nblk + i];
    a[j] = block_reduce(v, red, tid);
  }
  if (tid == 0) {
    float focal = a[0] * inv_n;
    float tprec = (a[2] + 1.0f) / (a[1] + 1.0f);
    float tsens = (a[4] + 1.0f) / (a[3] + 1.0f);
    float cld   = 1.0f - 2.0f * (tprec * tsens) / (tprec + tsens);
    out[0] = 0.9f * focal + 0.1f * cld;
  }
}

// ---------------- host launcher ----------------
extern "C" void kernel_launch(void* const* d_in, const int* in_sizes, int n_in,
                              void* d_out, int out_size, void* d_ws, size_t ws_size,
                              hipStream_t stream)
{
  const float* logits  = (const float*)d_in[0];
  const int*   targets = (const int*)d_in[1];
  float* out = (float*)d_out;
  float* ws  = (float*)d_ws;

  const int total = in_sizes[0];                 // 32*1*512*512
  const int batch = total / (IMG_H * IMG_W);
  dim3 grid(NTX, NTY, batch);
  const int nblk = NTX * NTY * batch;

  cl_dice_focal_tile_kernel<<<grid, NTPB, 0, stream>>>(logits, targets, ws);
  cl_dice_focal_final_kernel<<<1, NTPB, 0, stream>>>(ws, nblk,
                                                     1.0f / (float)total, out);
}